// GCN_74217034875214
// MI455X (gfx1250) — compile-verified
//
#include <hip/hip_runtime.h>

#define HF 128

typedef float v2f __attribute__((ext_vector_type(2)));
typedef float v8f __attribute__((ext_vector_type(8)));

__device__ __forceinline__ void atomicAddF(float* p, float v) {
    __hip_atomic_fetch_add(p, v, __ATOMIC_RELAXED, __HIP_MEMORY_SCOPE_AGENT);
}

// ---------------------------------------------------------------------------
// Zero-fill (float4 vectorized, tail-safe)
// ---------------------------------------------------------------------------
__global__ void gcn_zero_kernel(float* __restrict__ p, long n) {
    long i4 = ((long)blockIdx.x * blockDim.x + threadIdx.x) * 4;
    if (i4 + 3 < n) {
        float4 z; z.x = 0.f; z.y = 0.f; z.z = 0.f; z.w = 0.f;
        *(float4*)(p + i4) = z;
    } else {
        for (long j = i4; j < n; ++j) p[j] = 0.f;
    }
}

// ---------------------------------------------------------------------------
// Degree accumulation: outdeg[src[e]] += 1, indeg[dst[e]] += 1
// ---------------------------------------------------------------------------
__global__ void gcn_degree_kernel(const int* __restrict__ src,
                                  const int* __restrict__ dst,
                                  float* __restrict__ outdeg,
                                  float* __restrict__ indeg, int E) {
    int e = blockIdx.x * blockDim.x + threadIdx.x;
    if (e >= E) return;
    atomicAddF(&outdeg[src[e]], 1.0f);
    atomicAddF(&indeg[dst[e]], 1.0f);
}

// ---------------------------------------------------------------------------
// norm = (max(deg,1))^-0.5, in place on both degree arrays
// ---------------------------------------------------------------------------
__global__ void gcn_norm_kernel(float* __restrict__ a, float* __restrict__ b, int N) {
    int i = blockIdx.x * blockDim.x + threadIdx.x;
    if (i >= N) return;
    a[i] = rsqrtf(fmaxf(a[i], 1.0f));
    b[i] = rsqrtf(fmaxf(b[i], 1.0f));
}

// ---------------------------------------------------------------------------
// SpMM scatter: agg[dst[e], :] += X[src[e], :] * norm_src[src[e]]
// One wave (32 lanes) per edge; each lane handles 4 contiguous floats.
// X rows are L2-resident (51 MB node matrix << 192 MB L2).
// ---------------------------------------------------------------------------
__global__ void gcn_scatter_kernel(const float* __restrict__ X,
                                   const float* __restrict__ nsrc,
                                   const int* __restrict__ src,
                                   const int* __restrict__ dst,
                                   float* __restrict__ agg, int E) {
    int t = blockIdx.x * blockDim.x + threadIdx.x;
    int e = t >> 5;
    int lane = t & 31;
    if (e >= E) return;
    int s = src[e];
    int d = dst[e];
    float ns = nsrc[s];
    const float4 v = *(const float4*)(X + (long)s * HF + lane * 4);
    float* out = agg + (long)d * HF + lane * 4;
    atomicAddF(out + 0, v.x * ns);
    atomicAddF(out + 1, v.y * ns);
    atomicAddF(out + 2, v.z * ns);
    atomicAddF(out + 3, v.w * ns);
}

// ---------------------------------------------------------------------------
// Fused scale+GEMM+bias(+ReLU):  Y = act( (X * ndst[:,None]) @ W + b )
// fp32 WMMA 16x16x4. One wave per 16-row block, covering all 8 col tiles.
// A 16x4 f32 layout: lanes 0-15 -> K={0,1}, lanes 16-31 -> K={2,3} (VGPR0/1).
// C/D 16x16 f32: VGPR r holds M=r (lanes 0-15) / M=r+8 (lanes 16-31), N=lane&15.
// Safe in place (Y may alias X): each wave reads only its own 16 rows,
// all reads complete before the stores.
// ---------------------------------------------------------------------------
__global__ void gcn_gemm_kernel(const float* __restrict__ X,
                                const float* __restrict__ ndst,
                                const float* __restrict__ W,
                                const float* __restrict__ bias,
                                float* __restrict__ Y,
                                int Nn, int relu) {
    int wave = threadIdx.x >> 5;
    int lane = threadIdx.x & 31;
    int rowBlk = blockIdx.x * (blockDim.x >> 5) + wave;
    int m0 = rowBlk * 16;
    if (m0 >= Nn) return;            // wave-uniform: exact fit for N=100000

    int hs  = lane >> 4;             // 0: K pair {0,1}; 1: K pair {2,3}
    int l15 = lane & 15;
    long arow = m0 + l15;
    float scale = ndst[arow];
    const float* xr = X + arow * HF;

    v8f acc[8];
#pragma unroll
    for (int i = 0; i < 8; ++i) acc[i] = (v8f)0.0f;

    for (int k0 = 0; k0 < HF; k0 += 4) {
        int ka = k0 + hs * 2;
        v2f a;
        a.x = xr[ka]     * scale;
        a.y = xr[ka + 1] * scale;
        const float* w0 = W + (long)ka * HF;
#pragma unroll
        for (int nt = 0; nt < 8; ++nt) {
            int col = nt * 16 + l15;
            v2f b;
            b.x = w0[col];
            b.y = w0[HF + col];
            acc[nt] = __builtin_amdgcn_wmma_f32_16x16x4_f32(
                false, a, false, b, (short)0, acc[nt], false, false);
        }
    }

#pragma unroll
    for (int nt = 0; nt < 8; ++nt) {
        int col = nt * 16 + l15;
        float bb = bias[col];
#pragma unroll
        for (int r = 0; r < 8; ++r) {
            long row = m0 + hs * 8 + r;
            float v = acc[nt][r] + bb;
            if (relu) v = fmaxf(v, 0.0f);
            Y[row * HF + col] = v;
        }
    }
}

// ---------------------------------------------------------------------------
// Launcher
// ---------------------------------------------------------------------------
extern "C" void kernel_launch(void* const* d_in, const int* in_sizes, int n_in,
                              void* d_out, int out_size, void* d_ws, size_t ws_size,
                              hipStream_t stream) {
    (void)n_in; (void)out_size; (void)ws_size;
    const float* h  = (const float*)d_in[0];
    const float* W1 = (const float*)d_in[1];
    const float* b1 = (const float*)d_in[2];
    const float* W2 = (const float*)d_in[3];
    const float* b2 = (const float*)d_in[4];
    const int*   src = (const int*)d_in[5];
    const int*   dst = (const int*)d_in[6];

    const int Nn = in_sizes[0] / HF;
    const int E  = in_sizes[5];

    // Workspace layout: agg1[N*HF] | degO[N] | degI[N]   (~52 MB)
    float* agg1 = (float*)d_ws;
    float* degO = agg1 + (long)Nn * HF;
    float* degI = degO + Nn;
    float* out  = (float*)d_out;          // doubles as layer-2 aggregation buffer

    const long NH = (long)Nn * HF;

    { // zero workspace (agg1 + both degree arrays, contiguous)
        long n = NH + 2L * Nn;
        int blocks = (int)(((n + 3) / 4 + 255) / 256);
        gcn_zero_kernel<<<blocks, 256, 0, stream>>>(agg1, n);
    }
    { // zero d_out (layer-2 aggregation target)
        int blocks = (int)(((NH + 3) / 4 + 255) / 256);
        gcn_zero_kernel<<<blocks, 256, 0, stream>>>(out, NH);
    }

    gcn_degree_kernel<<<(E + 255) / 256, 256, 0, stream>>>(src, dst, degO, degI, E);
    gcn_norm_kernel<<<(Nn + 255) / 256, 256, 0, stream>>>(degO, degI, Nn);

    const int scatterBlocks = (int)(((long)E * 32 + 255) / 256);
    const int rowBlks = (Nn + 15) / 16;
    const int gemmBlocks = (rowBlks + 1) / 2;   // 2 waves (64 threads) per block

    // Layer 1: agg1 = scatter(h * nsrc); agg1 = relu(agg1*ndst @ W1 + b1) in place
    gcn_scatter_kernel<<<scatterBlocks, 256, 0, stream>>>(h, degO, src, dst, agg1, E);
    gcn_gemm_kernel<<<gemmBlocks, 64, 0, stream>>>(agg1, degI, W1, b1, agg1, Nn, 1);

    // Layer 2: out = scatter(h1 * nsrc); out = (out*ndst) @ W2 + b2 in place
    gcn_scatter_kernel<<<scatterBlocks, 256, 0, stream>>>(agg1, degO, src, dst, out, E);
    gcn_gemm_kernel<<<gemmBlocks, 64, 0, stream>>>(out, degI, W2, b2, out, Nn, 0);
}